// MultivariantLFQ_37520834298218
// MI455X (gfx1250) — compile-verified
//
#include <hip/hip_runtime.h>
#include <math.h>

typedef __attribute__((ext_vector_type(2))) float v2f;
typedef __attribute__((ext_vector_type(8))) float v8f;

#define NCODES   59049      // 3^10
#define NTILES   3691       // ceil(NCODES/16)
#define NPADCOL  59056
#define TOKEN    10
// logits computed in log2 domain: l2 = (2/T)*log2(e) * dot(z, code)
#define ASCALE   288.53900817779268f   // 200 * log2(e)
#define LN2      0.69314718055994531f
#define NEG_INF  (-3.0e38f)
#define LAST_NT  (NTILES - 1)          // 3690
#define LAST_W   (LAST_NT & 7)         // wave that owns the partial tile

// workspace layout (float offsets)
#define ROWMAX_OFF 0        // 1024 (log2-domain row max)
#define SE_OFF     1024     // 1024  sum 2^(l2 - max2)
#define SL_OFF     2048     // 1024  sum l2 * 2^(l2 - max2)
#define ACC_OFF    3072     // 1     commitment accumulator
#define COL_OFF    3080     // 59056 column prob sums

// ---------------------------------------------------------------------------
// ds_swizzle helpers (group-of-32 XOR patterns), EXEC all ones in our uses.
// ---------------------------------------------------------------------------
template <int PAT>
__device__ __forceinline__ float dswz(float x) {
  return __int_as_float(__builtin_amdgcn_ds_swizzle(__float_as_int(x), PAT));
}
#define SWAPX1  0x041F
#define SWAPX2  0x081F
#define SWAPX4  0x101F
#define SWAPX8  0x201F
#define SWAPX16 0x401F

__device__ __forceinline__ void bfly16_max(v8f& v) {
#pragma unroll
  for (int j = 0; j < 8; ++j) {
    v[j] = fmaxf(v[j], dswz<SWAPX1>(v[j]));
    v[j] = fmaxf(v[j], dswz<SWAPX2>(v[j]));
    v[j] = fmaxf(v[j], dswz<SWAPX4>(v[j]));
    v[j] = fmaxf(v[j], dswz<SWAPX8>(v[j]));
  }
}
__device__ __forceinline__ void bfly16_add(v8f& v) {
#pragma unroll
  for (int j = 0; j < 8; ++j) {
    v[j] += dswz<SWAPX1>(v[j]);
    v[j] += dswz<SWAPX2>(v[j]);
    v[j] += dswz<SWAPX4>(v[j]);
    v[j] += dswz<SWAPX8>(v[j]);
  }
}

// ---------------------------------------------------------------------------
// B operand for one 16-wide code tile: 6 ternary digits per lane, generated
// with a sequential divide chain. Lane-half selects starting digit (0 or 2).
// Digits at k>=10 are garbage but multiply zero A entries.
// ---------------------------------------------------------------------------
__device__ __forceinline__ void gen_b(int nbase, int lane, v2f* b) {
  int col = nbase + (lane & 15);
  int u = (lane & 16) ? (col / 9) : col;  // start at digit position koff
#pragma unroll
  for (int s = 0; s < 3; ++s) {
    int q0 = u / 3;
    b[s].x = (float)(u - 3 * q0 - 1);
    int q1 = q0 / 3;
    b[s].y = (float)(q0 - 3 * q1 - 1);
    u = q1 / 9;                           // skip 2 digit positions
  }
}

// A operand: 16 rows of zc for one M-tile, K padded 10->12, scale folded in.
__device__ __forceinline__ void load_a(const float* z, int mbase, int lane,
                                       v2f* a, float scale) {
  int row  = mbase + (lane & 15);
  int koff = (lane & 16) ? 2 : 0;
  int bb = row >> 8, hw = row & 255;
#pragma unroll
  for (int s = 0; s < 3; ++s) {
    int k0 = 4 * s + koff;
    a[s].x = (k0     < TOKEN) ? scale * z[bb * 2560 + k0 * 256 + hw] : 0.0f;
    a[s].y = (k0 + 1 < TOKEN) ? scale * z[bb * 2560 + (k0 + 1) * 256 + hw] : 0.0f;
  }
}

// Two 16x16 logit tiles (rows mbase..+15 and mbase+16..+31) sharing one B.
__device__ __forceinline__ void mm2(const v2f* a0, const v2f* a1, const v2f* b,
                                    v8f& c0, v8f& c1) {
  c0 = (v8f){};
  c1 = (v8f){};
#pragma unroll
  for (int s = 0; s < 3; ++s) {
    c0 = __builtin_amdgcn_wmma_f32_16x16x4_f32(false, a0[s], false, b[s],
                                               (short)0, c0, false, false);
    c1 = __builtin_amdgcn_wmma_f32_16x16x4_f32(false, a1[s], false, b[s],
                                               (short)0, c1, false, false);
  }
}

// ---------------------------------------------------------------------------
// Utility kernels
// ---------------------------------------------------------------------------
__global__ void MVLFQ_zero_ws(float* ws) {
  int i = blockIdx.x * 256 + threadIdx.x;
  if (i == 0) ws[ACC_OFF] = 0.0f;
  if (i < NPADCOL) ws[COL_OFF + i] = 0.0f;
}

__global__ void MVLFQ_quantize(const float* __restrict__ z,
                               float* __restrict__ out,
                               float* __restrict__ idx_out,
                               float* __restrict__ commit_acc) {
  int m = blockIdx.x * 256 + threadIdx.x;
  if (m >= 1024) return;
  int bb = m >> 8, hw = m & 255;
  float csum = 0.0f;
  int idx = 0, p3 = 1;
#pragma unroll
  for (int c = 0; c < TOKEN; ++c) {
    float zv   = z[bb * 2560 + c * 256 + hw];
    float bq   = fminf(fmaxf(rintf(zv + 1.0f), 0.0f), 2.0f);
    float zsq  = bq - 1.0f;
    out[bb * 2560 + c * 256 + hw] = zsq;     // STE forward value
    float d = zsq - zv;
    csum += d * d;
    idx += (int)bq * p3;
    p3 *= 3;
  }
  idx_out[m] = (float)idx;
  atomicAdd(commit_acc, csum);
}

// ---------------------------------------------------------------------------
// Pass 1: per-row max of log2-domain logits.
// ---------------------------------------------------------------------------
__global__ void MVLFQ_rowmax(const float* __restrict__ z,
                             float* __restrict__ rowmax) {
  int lane = threadIdx.x & 31;
  int wave = threadIdx.x >> 5;
  int mbase = blockIdx.x * 32;

  v2f a0[3], a1[3];
  load_a(z, mbase,      lane, a0, ASCALE);
  load_a(z, mbase + 16, lane, a1, ASCALE);

  v8f best0, best1;
#pragma unroll
  for (int j = 0; j < 8; ++j) { best0[j] = NEG_INF; best1[j] = NEG_INF; }

  for (int nt = wave; nt < LAST_NT; nt += 8) {   // full tiles, no masking
    v2f b[3];
    gen_b(nt * 16, lane, b);
    v8f c0, c1;
    mm2(a0, a1, b, c0, c1);
#pragma unroll
    for (int j = 0; j < 8; ++j) {
      best0[j] = fmaxf(best0[j], c0[j]);
      best1[j] = fmaxf(best1[j], c1[j]);
    }
  }
  if (wave == LAST_W) {                          // peeled partial tile
    v2f b[3];
    gen_b(LAST_NT * 16, lane, b);
    v8f c0, c1;
    mm2(a0, a1, b, c0, c1);
    bool valid = (LAST_NT * 16 + (lane & 15)) < NCODES;
#pragma unroll
    for (int j = 0; j < 8; ++j) {
      best0[j] = fmaxf(best0[j], valid ? c0[j] : NEG_INF);
      best1[j] = fmaxf(best1[j], valid ? c1[j] : NEG_INF);
    }
  }
  bfly16_max(best0);
  bfly16_max(best1);

  __shared__ float red[8][32];
  if (lane == 0)
    for (int j = 0; j < 8; ++j) { red[wave][j] = best0[j]; red[wave][16 + j] = best1[j]; }
  if (lane == 16)
    for (int j = 0; j < 8; ++j) { red[wave][8 + j] = best0[j]; red[wave][24 + j] = best1[j]; }
  __syncthreads();
  if (threadIdx.x < 32) {
    float m = red[0][threadIdx.x];
    for (int w = 1; w < 8; ++w) m = fmaxf(m, red[w][threadIdx.x]);
    rowmax[mbase + threadIdx.x] = m;
  }
}

// ---------------------------------------------------------------------------
// Pass 2: per-row sum 2^(l2-max) and sum l2*2^(l2-max).
// ---------------------------------------------------------------------------
__global__ void MVLFQ_rowsums(const float* __restrict__ z,
                              const float* __restrict__ rowmax,
                              float* __restrict__ se_out,
                              float* __restrict__ sl_out) {
  int lane = threadIdx.x & 31;
  int wave = threadIdx.x >> 5;
  int mbase = blockIdx.x * 32;
  int half  = (lane & 16) ? 8 : 0;

  v2f a0[3], a1[3];
  load_a(z, mbase,      lane, a0, ASCALE);
  load_a(z, mbase + 16, lane, a1, ASCALE);

  float rm0[8], rm1[8];
#pragma unroll
  for (int j = 0; j < 8; ++j) {
    rm0[j] = rowmax[mbase + half + j];
    rm1[j] = rowmax[mbase + 16 + half + j];
  }

  v8f se0 = {}, sl0 = {}, se1 = {}, sl1 = {};
  for (int nt = wave; nt < LAST_NT; nt += 8) {
    v2f b[3];
    gen_b(nt * 16, lane, b);
    v8f c0, c1;
    mm2(a0, a1, b, c0, c1);
#pragma unroll
    for (int j = 0; j < 8; ++j) {
      float e0 = __builtin_amdgcn_exp2f(c0[j] - rm0[j]);
      float e1 = __builtin_amdgcn_exp2f(c1[j] - rm1[j]);
      se0[j] += e0; sl0[j] += e0 * c0[j];
      se1[j] += e1; sl1[j] += e1 * c1[j];
    }
  }
  if (wave == LAST_W) {
    v2f b[3];
    gen_b(LAST_NT * 16, lane, b);
    v8f c0, c1;
    mm2(a0, a1, b, c0, c1);
    bool valid = (LAST_NT * 16 + (lane & 15)) < NCODES;
#pragma unroll
    for (int j = 0; j < 8; ++j) {
      float e0 = valid ? __builtin_amdgcn_exp2f(c0[j] - rm0[j]) : 0.0f;
      float e1 = valid ? __builtin_amdgcn_exp2f(c1[j] - rm1[j]) : 0.0f;
      se0[j] += e0; sl0[j] += e0 * c0[j];
      se1[j] += e1; sl1[j] += e1 * c1[j];
    }
  }
  bfly16_add(se0); bfly16_add(sl0);
  bfly16_add(se1); bfly16_add(sl1);

  __shared__ float red_se[8][32];
  __shared__ float red_sl[8][32];
  if (lane == 0)
    for (int j = 0; j < 8; ++j) {
      red_se[wave][j] = se0[j]; red_se[wave][16 + j] = se1[j];
      red_sl[wave][j] = sl0[j]; red_sl[wave][16 + j] = sl1[j];
    }
  if (lane == 16)
    for (int j = 0; j < 8; ++j) {
      red_se[wave][8 + j] = se0[j]; red_se[wave][24 + j] = se1[j];
      red_sl[wave][8 + j] = sl0[j]; red_sl[wave][24 + j] = sl1[j];
    }
  __syncthreads();
  if (threadIdx.x < 32) {
    float tse = 0.0f, tsl = 0.0f;
    for (int w = 0; w < 8; ++w) { tse += red_se[w][threadIdx.x]; tsl += red_sl[w][threadIdx.x]; }
    se_out[mbase + threadIdx.x] = tse;
    sl_out[mbase + threadIdx.x] = tsl;
  }
}

// ---------------------------------------------------------------------------
// Pass 3: column prob sums (avg_probs numerator) via f32 global atomics.
// ---------------------------------------------------------------------------
__global__ void MVLFQ_colsums(const float* __restrict__ z,
                              const float* __restrict__ rowmax,
                              const float* __restrict__ se_in,
                              float* __restrict__ colsum) {
  int lane = threadIdx.x & 31;
  int wave = threadIdx.x >> 5;
  int mbase = blockIdx.x * 32;
  int half  = (lane & 16) ? 8 : 0;

  v2f a0[3], a1[3];
  load_a(z, mbase,      lane, a0, ASCALE);
  load_a(z, mbase + 16, lane, a1, ASCALE);

  float rm0[8], rm1[8], is0[8], is1[8];
#pragma unroll
  for (int j = 0; j < 8; ++j) {
    rm0[j] = rowmax[mbase + half + j];
    rm1[j] = rowmax[mbase + 16 + half + j];
    is0[j] = 1.0f / se_in[mbase + half + j];
    is1[j] = 1.0f / se_in[mbase + 16 + half + j];
  }

  for (int nt = wave; nt < LAST_NT; nt += 8) {
    v2f b[3];
    gen_b(nt * 16, lane, b);
    v8f c0, c1;
    mm2(a0, a1, b, c0, c1);
    float psum = 0.0f;
#pragma unroll
    for (int j = 0; j < 8; ++j) {
      psum += __builtin_amdgcn_exp2f(c0[j] - rm0[j]) * is0[j];
      psum += __builtin_amdgcn_exp2f(c1[j] - rm1[j]) * is1[j];
    }
    psum += dswz<SWAPX16>(psum);        // rows 0-7,16-23 + rows 8-15,24-31 of column
    if (lane < 16) atomicAdd(&colsum[nt * 16 + lane], psum);
  }
  if (wave == LAST_W) {
    v2f b[3];
    gen_b(LAST_NT * 16, lane, b);
    v8f c0, c1;
    mm2(a0, a1, b, c0, c1);
    float psum = 0.0f;
#pragma unroll
    for (int j = 0; j < 8; ++j) {
      psum += __builtin_amdgcn_exp2f(c0[j] - rm0[j]) * is0[j];
      psum += __builtin_amdgcn_exp2f(c1[j] - rm1[j]) * is1[j];
    }
    psum += dswz<SWAPX16>(psum);
    int col = LAST_NT * 16 + lane;
    if (lane < 16 && col < NCODES) atomicAdd(&colsum[col], psum);
  }
}

// ---------------------------------------------------------------------------
// Finalize: entropies + loss scalars (log2 domain -> nats via LN2).
// ---------------------------------------------------------------------------
__global__ void MVLFQ_finalize(const float* __restrict__ rowmax,
                               const float* __restrict__ se,
                               const float* __restrict__ sl,
                               const float* __restrict__ colsum,
                               const float* __restrict__ commit_acc,
                               float* __restrict__ out_scalars) {
  __shared__ float sred[256];
  int tid = threadIdx.x;

  float hsum = 0.0f;   // sum of per-row entropies (nats)
  for (int m = tid; m < 1024; m += 256) {
    float s = se[m];
    hsum += LN2 * (rowmax[m] + __builtin_amdgcn_logf(s) - sl[m] / s);
  }
  float asum = 0.0f;   // -sum avg_p * ln(avg_p + 1e-5)
  for (int n = tid; n < NCODES; n += 256) {
    float p = colsum[n] * (1.0f / 1024.0f);
    asum -= p * (LN2 * __builtin_amdgcn_logf(p + 1e-5f));
  }

  sred[tid] = hsum; __syncthreads();
  for (int s = 128; s > 0; s >>= 1) { if (tid < s) sred[tid] += sred[tid + s]; __syncthreads(); }
  float hsum_tot = sred[0]; __syncthreads();

  sred[tid] = asum; __syncthreads();
  for (int s = 128; s > 0; s >>= 1) { if (tid < s) sred[tid] += sred[tid + s]; __syncthreads(); }
  float asum_tot = sred[0];

  if (tid == 0) {
    float pse    = hsum_tot * (1.0f / 1024.0f);
    float ae     = asum_tot;
    float entl   = 0.1f * (pse - ae);
    float commit = 0.25f * commit_acc[0] * (1.0f / 10240.0f);
    out_scalars[0] = commit + entl;   // loss
    out_scalars[1] = commit;          // commitment_loss
    out_scalars[2] = entl;            // entropy_loss
    out_scalars[3] = pse;             // per_sample_entropy
    out_scalars[4] = ae;              // avg_entropy
  }
}

// ---------------------------------------------------------------------------
extern "C" void kernel_launch(void* const* d_in, const int* in_sizes, int n_in,
                              void* d_out, int out_size, void* d_ws, size_t ws_size,
                              hipStream_t stream) {
  const float* z = (const float*)d_in[0];
  float* out = (float*)d_out;
  float* ws  = (float*)d_ws;

  // d_out: [0,10240) z_ste (b,c,h,w) | [10240,10245) scalars |
  //        [10245,11269) min_encoding_indices (as float)
  MVLFQ_zero_ws<<<(NPADCOL + 255) / 256, 256, 0, stream>>>(ws);
  MVLFQ_quantize<<<4, 256, 0, stream>>>(z, out, out + 10245, ws + ACC_OFF);
  MVLFQ_rowmax <<<32, 256, 0, stream>>>(z, ws + ROWMAX_OFF);
  MVLFQ_rowsums<<<32, 256, 0, stream>>>(z, ws + ROWMAX_OFF, ws + SE_OFF, ws + SL_OFF);
  MVLFQ_colsums<<<32, 256, 0, stream>>>(z, ws + ROWMAX_OFF, ws + SE_OFF, ws + COL_OFF);
  MVLFQ_finalize<<<1, 256, 0, stream>>>(ws + ROWMAX_OFF, ws + SE_OFF, ws + SL_OFF,
                                        ws + COL_OFF, ws + ACC_OFF, out + 10240);
}